// GraphLevelGNN_87144886435840
// MI455X (gfx1250) — compile-verified
//
#include <hip/hip_runtime.h>

typedef __attribute__((ext_vector_type(2))) float v2f;
typedef __attribute__((ext_vector_type(8))) float v8f;

#define GCN_N 100000

// ---------------- degree / dinv ----------------

__global__ void zero_f32_kernel(float* __restrict__ p, int n) {
    int i = blockIdx.x * blockDim.x + threadIdx.x;
    if (i < n) p[i] = 0.0f;
}

__global__ void deg_count_kernel(const int* __restrict__ dst, float* __restrict__ deg, int E) {
    int e = blockIdx.x * blockDim.x + threadIdx.x;
    if (e < E) atomicAdd(&deg[dst[e]], 1.0f);
}

__global__ void dinv_kernel(float* __restrict__ deg, int n) {
    int i = blockIdx.x * blockDim.x + threadIdx.x;
    if (i < n) {
        float d = deg[i] + 1.0f;      // self-loop
        deg[i] = 1.0f / sqrtf(d);     // in-place: deg -> dinv
    }
}

// ---------------- W packing: fragment-order, zero-padded ----------------
// Bp float index f = kt*4*CP + half*2*CP + 2*col + v  <=>  v2f element (kt*2+half)*CP + col
__global__ void pack_w_kernel(const float* __restrict__ W, float* __restrict__ Bp,
                              int K, int C, int total4CP, int CP) {
    int f = blockIdx.x * blockDim.x + threadIdx.x;
    if (f >= total4CP) return;
    int kt   = f / (4 * CP);
    int r    = f - kt * 4 * CP;
    int half = r / (2 * CP);
    int r2   = r - half * 2 * CP;
    int col  = r2 >> 1;
    int v    = r2 & 1;
    int k    = kt * 4 + half * 2 + v;
    Bp[f] = (k < K && col < C) ? W[k * C + col] : 0.0f;
}

// ---------------- GEMM: out[N x CP] = X[N x K] @ Bp, fp32 WMMA ----------------
// One wave per 16x16 tile, 4 waves/block, tn-major tile order (B reuse in L0).
// No bounds checks: K, CP compile-time; N % 16 == 0; padding columns are zero.
template <int K, int CP>
__global__ void gemm_wmma_kernel(const float* __restrict__ X, const float* __restrict__ Bp,
                                 float* __restrict__ out, int tilesM) {
    const int wave = threadIdx.x >> 5;
    const int tile = blockIdx.x * 4 + wave;
    if (tile >= tilesM * (CP / 16)) return;
    const int tn = tile / tilesM;          // tn-major: consecutive waves share tn
    const int tm = tile - tn * tilesM;
    const int lane = threadIdx.x & 31;
    const int half = lane >> 4;            // 0: lanes 0-15, 1: lanes 16-31
    const int mn   = lane & 15;

    // A frag: lane (mn,half) vgpr v -> X[tm*16+mn][kt*4 + half*2 + v]  (8B aligned)
    const float* aPtr = X  + (size_t)(tm * 16 + mn) * K + half * 2;
    // B frag: packed v2f at (kt*2+half)*CP + (tn*16+mn)
    const float* bPtr = Bp + 2 * (size_t)(half * CP + tn * 16 + mn);

    v8f acc = {0.0f, 0.0f, 0.0f, 0.0f, 0.0f, 0.0f, 0.0f, 0.0f};
#pragma unroll 4
    for (int kt = 0; kt < K / 4; ++kt) {
        v2f a = *(const v2f*)(aPtr + kt * 4);
        v2f b = *(const v2f*)(bPtr + kt * 4 * CP);
        acc = __builtin_amdgcn_wmma_f32_16x16x4_f32(false, a, false, b,
                                                    (short)0, acc, false, false);
    }

    // C/D: vgpr j -> row tm*16 + half*8 + j, col tn*16 + mn
    float* o = out + (size_t)(tm * 16 + half * 8) * CP + (tn * 16 + mn);
#pragma unroll
    for (int j = 0; j < 8; ++j)
        o[(size_t)j * CP] = acc[j];
}

// ---------------- self-loop + bias init (keeps pad columns zero) ----------------
__global__ void self_bias_kernel(const float* __restrict__ xw, const float* __restrict__ dinv,
                                 const float* __restrict__ bias, float* __restrict__ h,
                                 int N, int C, int CS) {
    long long tid   = (long long)blockIdx.x * blockDim.x + threadIdx.x;
    long long total = (long long)N * CS;
    if (tid < total) {
        int i = (int)(tid / CS);
        int c = (int)(tid - (long long)i * CS);
        float di = dinv[i];
        float b  = (c < C) ? bias[c] : 0.0f;   // xw pad cols are exactly 0
        h[tid] = xw[tid] * di * di + b;
    }
}

// ---------------- edge scatter: one wave per edge, L2 f32 atomics ----------------
__global__ void edge_scatter_kernel(const int* __restrict__ ei, const float* __restrict__ dinv,
                                    const float* __restrict__ xw, float* __restrict__ h,
                                    int E, int C, int CS) {
    int e = blockIdx.x * (blockDim.x >> 5) + (threadIdx.x >> 5);
    if (e >= E) return;
    int lane = threadIdx.x & 31;
    int s = ei[e];
    int d = ei[E + e];
    float nrm = dinv[s] * dinv[d];
    const float* xr = xw + (size_t)s * CS;
    float*       hr = h  + (size_t)d * CS;
    for (int c = lane; c < C; c += 32)
        atomicAdd(&hr[c], xr[c] * nrm);
}

__global__ void edge_scatter_c1_kernel(const int* __restrict__ ei, const float* __restrict__ dinv,
                                       const float* __restrict__ xw, float* __restrict__ out,
                                       int E, int CS) {
    int e = blockIdx.x * blockDim.x + threadIdx.x;
    if (e < E) {
        int s = ei[e];
        int d = ei[E + e];
        atomicAdd(&out[d], xw[(size_t)s * CS] * dinv[s] * dinv[d]);
    }
}

__global__ void out_init_kernel(const float* __restrict__ xw, const float* __restrict__ dinv,
                                const float* __restrict__ b3, float* __restrict__ out,
                                int N, int CS) {
    int i = blockIdx.x * blockDim.x + threadIdx.x;
    if (i < N) {
        float di = dinv[i];
        out[i] = xw[(size_t)i * CS] * di * di + b3[0];
    }
}

__global__ void relu_kernel(float* __restrict__ h, long long n) {
    long long i = (long long)blockIdx.x * blockDim.x + threadIdx.x;
    if (i < n) h[i] = fmaxf(h[i], 0.0f);
}

// ---------------- launch ----------------

extern "C" void kernel_launch(void* const* d_in, const int* in_sizes, int n_in,
                              void* d_out, int out_size, void* d_ws, size_t ws_size,
                              hipStream_t stream) {
    const float* x  = (const float*)d_in[0];
    const int*   ei = (const int*)d_in[1];
    const float* W1 = (const float*)d_in[2];
    const float* b1 = (const float*)d_in[3];
    const float* W2 = (const float*)d_in[4];
    const float* b2 = (const float*)d_in[5];
    const float* W3 = (const float*)d_in[6];
    const float* b3 = (const float*)d_in[7];
    float* out = (float*)d_out;

    const int N  = GCN_N;
    const int E  = in_sizes[1] / 2;               // edge_index is [2, E]
    const int IC = 128, H1 = 71, H2 = 82;
    const int CP1 = 80, CP2 = 96, CP3 = 16;       // padded widths (mult of 16)

    // workspace: dinv [N] | xw [N*96] | h [N*96] | Wp [<=128*80]
    float* dinv = (float*)d_ws;
    float* xw   = dinv + N;
    float* h    = xw + (size_t)N * CP2;
    float* Wp   = h  + (size_t)N * CP2;

    const int B = 256;
    const int tilesM = N / 16;                    // 6250, exact
    long long tot;

    // degrees -> dinv
    zero_f32_kernel<<<(N + B - 1) / B, B, 0, stream>>>(dinv, N);
    deg_count_kernel<<<(E + B - 1) / B, B, 0, stream>>>(ei + E, dinv, E);
    dinv_kernel<<<(N + B - 1) / B, B, 0, stream>>>(dinv, N);

    // ---- layer 1: x @ W1 -> xw[N x 80], aggregate -> h, relu ----
    {
        int totalW = (IC / 4) * 4 * CP1;          // 128 x 80
        pack_w_kernel<<<(totalW + B - 1) / B, B, 0, stream>>>(W1, Wp, IC, H1, totalW, CP1);
        int tiles = tilesM * (CP1 / 16);
        gemm_wmma_kernel<128, CP1><<<(tiles + 3) / 4, 128, 0, stream>>>(x, Wp, xw, tilesM);
        tot = (long long)N * CP1;
        self_bias_kernel<<<(int)((tot + B - 1) / B), B, 0, stream>>>(xw, dinv, b1, h, N, H1, CP1);
        edge_scatter_kernel<<<(E + 7) / 8, B, 0, stream>>>(ei, dinv, xw, h, E, H1, CP1);
        relu_kernel<<<(int)((tot + B - 1) / B), B, 0, stream>>>(h, tot);
    }

    // ---- layer 2: h[N x 80] @ W2 -> xw[N x 96], aggregate -> h (no relu) ----
    {
        int totalW = (CP1 / 4) * 4 * CP2;         // 80 x 96 (rows 71.. zero)
        pack_w_kernel<<<(totalW + B - 1) / B, B, 0, stream>>>(W2, Wp, H1, H2, totalW, CP2);
        int tiles = tilesM * (CP2 / 16);
        gemm_wmma_kernel<CP1, CP2><<<(tiles + 3) / 4, 128, 0, stream>>>(h, Wp, xw, tilesM);
        tot = (long long)N * CP2;
        self_bias_kernel<<<(int)((tot + B - 1) / B), B, 0, stream>>>(xw, dinv, b2, h, N, H2, CP2);
        edge_scatter_kernel<<<(E + 7) / 8, B, 0, stream>>>(ei, dinv, xw, h, E, H2, CP2);
    }

    // ---- layer 3: h[N x 96] @ W3 -> xw[N x 16], aggregate into d_out ----
    {
        int totalW = (CP2 / 4) * 4 * CP3;         // 96 x 16 (rows 82.., cols 1.. zero)
        pack_w_kernel<<<(totalW + B - 1) / B, B, 0, stream>>>(W3, Wp, H2, 1, totalW, CP3);
        int tiles = tilesM * (CP3 / 16);
        gemm_wmma_kernel<CP2, CP3><<<(tiles + 3) / 4, 128, 0, stream>>>(h, Wp, xw, tilesM);
        out_init_kernel<<<(N + B - 1) / B, B, 0, stream>>>(xw, dinv, b3, out, N, CP3);
        edge_scatter_c1_kernel<<<(E + B - 1) / B, B, 0, stream>>>(ei, dinv, xw, out, E, CP3);
    }
}